// MEPG_Loss_33097017983361
// MI455X (gfx1250) — compile-verified
//
#include <hip/hip_runtime.h>
#include <hip/hip_bf16.h>
#include <math.h>

// MEPG loss, collapsed to per-simulation sums (no cumsum materialization):
//   out = (1/(T*S)) * sum_s W_s * LL_s
//   LL_s = sum_t ll(t,s)
//   W_s  = sum_t (t+1)*(exp(r)-alpha*ll) - sum_t baseline(t,s)
//
// Layer 1 of both MLPs fused (128 hidden rows) on V_WMMA_F32_16X16X4_F32:
//   H^T[128x16] = W1aug[128x8(K pad)] x X^T[8x16-samples]
// A (weights) register-resident in A-layout; B (states) loaded per batch;
// C layout keeps sample = lane%16 invariant, so layer 2 is per-lane FMA
// against LDS-broadcast W2 rows + one shfl_xor(16).
//
// v3: (1) '#pragma unroll 1' on the t-batch loop (v2's 4x unroll caused
// scratch spills + vgpr-msb churn); (2) loop-invariant Cholesky/bias scalars
// pinned via opaque asm so the compiler can't rematerialize divisions inside
// the loop; (3) LDS weight loads kept in-loop via opaque zero offset.

typedef __attribute__((ext_vector_type(2))) float v2f;
typedef __attribute__((ext_vector_type(4))) float v4f;
typedef __attribute__((ext_vector_type(8))) float v8f;

#define N_SIMS 4096
#define T_LEN  500
#define D_DIM  6
#define A_DIM  3
#define H_DIM  64     // per-MLP hidden
#define HT     128    // fused hidden: policy 64 + baseline 64
#define ALPHA  0.1f

__global__ __launch_bounds__(256) void mepg_main(
    const float* __restrict__ st,   // [S, D, T]
    const float* __restrict__ at,   // [S, A, T]
    const float* __restrict__ rw,   // [S, T]
    const float* __restrict__ pW1, const float* __restrict__ pb1,   // 64x6, 64
    const float* __restrict__ pW2, const float* __restrict__ pb2,   // 3x64, 3
    const float* __restrict__ bW1, const float* __restrict__ bb1,   // 64x6, 64
    const float* __restrict__ bW2, const float* __restrict__ bb2,   // 1x64, 1
    const float* __restrict__ sd,                                   // 3x3
    float* __restrict__ partials)                                   // [S]
{
  __shared__ v4f  w2s[HT];        // W2aug[j] = {pol_a0, pol_a1, pol_a2, baseline}
  __shared__ float red[8][2];

  const int s    = blockIdx.x;
  const int tid  = threadIdx.x;
  const int lane = tid & 31;
  const int wv   = tid >> 5;      // 8 waves per block
  const int m    = lane & 15;     // sample column (N dim of WMMA)
  const int hb   = lane >> 4;     // lane half

  // ---- stage fused second-layer weights W2aug[j][a] into LDS ----
  if (tid < HT) {
    const int j = tid;
    v4f w;
    if (j < H_DIM) {
      w.x = pW2[0 * H_DIM + j]; w.y = pW2[1 * H_DIM + j];
      w.z = pW2[2 * H_DIM + j]; w.w = 0.f;
    } else {
      w.x = 0.f; w.y = 0.f; w.z = 0.f; w.w = bW2[j - H_DIM];
    }
    w2s[j] = w;
  }
  __syncthreads();

  // ---- preload A = W1aug[128 x 8] in WMMA A-layout (constant, 32 VGPRs) ----
  // A layout (32-bit 16x4): lanes 0-15 hold M=row; VGPR0: K=k0 (low half) /
  // k0+2 (high half); VGPR1: K=k0+1 / k0+3.
  // W1aug[j][k]: k<6 -> W1[j,k]; k==6 -> b1[j]; k==7 -> 0 (pad).
  v2f Aw[8][2];
#pragma unroll
  for (int nt = 0; nt < 8; ++nt) {
    const int j = nt * 16 + m;
    const float* W1  = (j < H_DIM) ? pW1 : bW1;
    const float* B1p = (j < H_DIM) ? pb1 : bb1;
    const int jr = (j < H_DIM) ? j : (j - H_DIM);
#pragma unroll
    for (int c = 0; c < 2; ++c) {
      const int k0 = 4 * c + 2 * hb;
      const int k1 = k0 + 1;
      float a0 = (k0 < 6) ? W1[jr * D_DIM + k0] : ((k0 == 6) ? B1p[jr] : 0.f);
      float a1 = (k1 < 6) ? W1[jr * D_DIM + k1] : ((k1 == 6) ? B1p[jr] : 0.f);
      v2f v; v.x = a0; v.y = a1;
      Aw[nt][c] = v;
    }
  }

  // ---- 3x3 Cholesky of sd, lower-triangular inverse, log-det constant ----
  const float s00 = sd[0], s10 = sd[3], s11 = sd[4];
  const float s20 = sd[6], s21 = sd[7], s22 = sd[8];
  const float L00 = sqrtf(s00);
  const float L10 = s10 / L00;
  const float L11 = sqrtf(s11 - L10 * L10);
  const float L20 = s20 / L00;
  const float L21 = (s21 - L20 * L10) / L11;
  const float L22 = sqrtf(s22 - L20 * L20 - L21 * L21);
  float Li00 = 1.f / L00, Li11 = 1.f / L11, Li22 = 1.f / L22;
  float Li10 = -L10 * Li00 * Li11;
  float Li21 = -L21 * Li11 * Li22;
  float Li20 = (L10 * L21 - L11 * L20) * Li00 * Li11 * Li22;
  float llconst = -(logf(L00) + logf(L11) + logf(L22))
                  - 0.5f * (float)A_DIM * logf(6.283185307179586f);
  float b2r0 = pb2[0], b2r1 = pb2[1], b2r2 = pb2[2], b2r3 = bb2[0];

  // Pin the invariants as opaque register values: prevents the compiler from
  // rematerializing the division/log sequences inside the hot loop.
  asm volatile("" : "+v"(Li00), "+v"(Li10), "+v"(Li11), "+v"(Li20),
                    "+v"(Li21), "+v"(Li22), "+v"(llconst),
                    "+v"(b2r0), "+v"(b2r1), "+v"(b2r2), "+v"(b2r3));

  const float* stS = st + (size_t)s * D_DIM * T_LEN;
  const float* atS = at + (size_t)s * A_DIM * T_LEN;
  const float* rwS = rw + (size_t)s * T_LEN;

  float accLL = 0.f, accW = 0.f;
  const float inf = __builtin_inff();

  // 8 waves x 4 iters x 16 samples = 512 slots >= T=500 (tail masked)
#pragma unroll 1
  for (int it = 0; it < 4; ++it) {
    const int t  = (wv * 4 + it) * 16 + m;
    const int tc = (t < T_LEN) ? t : (T_LEN - 1);

    // Opaque zero: keeps the LDS weight loads inside the loop (no hoisting
    // into 256 hoisted VGPRs).
    int zoff = 0;
    asm volatile("" : "+s"(zoff));

    // ---- B = X^T tiles (B layout symmetric with A/C half-split):
    // VGPR0: K = 2*hb, VGPR1: K = 2*hb+1 (chunk k0=0); chunk k0=4 likewise.
    const int kb = 2 * hb;
    v2f B0; B0.x = stS[(kb + 0) * T_LEN + tc];
            B0.y = stS[(kb + 1) * T_LEN + tc];
    const float x4 = stS[4 * T_LEN + tc];
    const float x5 = stS[5 * T_LEN + tc];
    v2f B1; B1.x = hb ? 1.f : x4;   // k=6 is the bias column (x=1), k=7 pad (0)
            B1.y = hb ? 0.f : x5;

    float P0 = 0.f, P1 = 0.f, P2 = 0.f, P3 = 0.f;
#pragma unroll
    for (int nt = 0; nt < 8; ++nt) {
      v8f c = {0.f, 0.f, 0.f, 0.f, 0.f, 0.f, 0.f, 0.f};
      c = __builtin_amdgcn_wmma_f32_16x16x4_f32(false, Aw[nt][0], false, B0,
                                                (short)0, c, false, false);
      c = __builtin_amdgcn_wmma_f32_16x16x4_f32(false, Aw[nt][1], false, B1,
                                                (short)0, c, false, false);
      // C layout: VGPR v, lane l -> (hidden j = nt*16 + v + 8*hb, sample l%16)
#pragma unroll
      for (int v = 0; v < 8; ++v) {
        const float h = __builtin_amdgcn_fmed3f(c[v], 0.f, inf);  // 1-op ReLU
        const v4f w4 = w2s[nt * 16 + v + 8 * hb + zoff];  // LDS half-broadcast
        P0 = fmaf(w4.x, h, P0);
        P1 = fmaf(w4.y, h, P1);
        P2 = fmaf(w4.z, h, P2);
        P3 = fmaf(w4.w, h, P3);
      }
    }
    // join the two K-halves (hb=0 covered j%16 in 0..7, hb=1 covered 8..15)
    P0 += __shfl_xor(P0, 16, 32);
    P1 += __shfl_xor(P1, 16, 32);
    P2 += __shfl_xor(P2, 16, 32);
    P3 += __shfl_xor(P3, 16, 32);

    // ---- per-sample tail (every lane computes; duplicates/tail masked) ----
    const float mu0 = P0 + b2r0, mu1 = P1 + b2r1, mu2 = P2 + b2r2;
    const float base = P3 + b2r3;
    const float r0 = atS[0 * T_LEN + tc] - mu0;
    const float r1 = atS[1 * T_LEN + tc] - mu1;
    const float r2 = atS[2 * T_LEN + tc] - mu2;
    const float z0 = Li00 * r0;
    const float z1 = Li10 * r0 + Li11 * r1;
    const float z2 = Li20 * r0 + Li21 * r1 + Li22 * r2;
    const float ll = -0.5f * (z0 * z0 + z1 * z1 + z2 * z2) + llconst;
    const float g  = expf(rwS[tc]) - ALPHA * ll;
    const float sel = (t < T_LEN && hb == 0) ? 1.f : 0.f;
    accLL += sel * ll;
    accW  += sel * ((float)(t + 1) * g - base);
  }

  // ---- reduce within wave (lanes 16-31 carry zeros), then across waves ----
#pragma unroll
  for (int off = 1; off < 16; off <<= 1) {
    accLL += __shfl_xor(accLL, off, 32);
    accW  += __shfl_xor(accW,  off, 32);
  }
  if (lane == 0) { red[wv][0] = accLL; red[wv][1] = accW; }
  __syncthreads();
  if (tid == 0) {
    float LL = 0.f, W = 0.f;
    for (int w = 0; w < 8; ++w) { LL += red[w][0]; W += red[w][1]; }
    partials[s] = W * LL;   // fully overwritten every call -> deterministic
  }
}

__global__ __launch_bounds__(256) void mepg_reduce(const float* __restrict__ partials,
                                                   float* __restrict__ out) {
  __shared__ float sm[256];
  const int tid = threadIdx.x;
  float a = 0.f;
  for (int i = tid; i < N_SIMS; i += 256) a += partials[i];   // fixed order
  sm[tid] = a;
  __syncthreads();
  for (int stp = 128; stp > 0; stp >>= 1) {
    if (tid < stp) sm[tid] += sm[tid + stp];
    __syncthreads();
  }
  if (tid == 0) out[0] = sm[0] / ((float)T_LEN * (float)N_SIMS);
}

extern "C" void kernel_launch(void* const* d_in, const int* in_sizes, int n_in,
                              void* d_out, int out_size, void* d_ws, size_t ws_size,
                              hipStream_t stream) {
  const float* st  = (const float*)d_in[0];
  const float* at  = (const float*)d_in[1];
  const float* rw  = (const float*)d_in[2];
  const float* pW1 = (const float*)d_in[3];
  const float* pb1 = (const float*)d_in[4];
  const float* pW2 = (const float*)d_in[5];
  const float* pb2 = (const float*)d_in[6];
  const float* bW1 = (const float*)d_in[7];
  const float* bb1 = (const float*)d_in[8];
  const float* bW2 = (const float*)d_in[9];
  const float* bb2 = (const float*)d_in[10];
  const float* sd  = (const float*)d_in[11];
  float* partials  = (float*)d_ws;          // 4096 floats = 16 KB
  float* out       = (float*)d_out;

  mepg_main<<<N_SIMS, 256, 0, stream>>>(st, at, rw, pW1, pb1, pW2, pb2,
                                        bW1, bb1, bW2, bb2, sd, partials);
  mepg_reduce<<<1, 256, 0, stream>>>(partials, out);
}